// SegmentedPolynomialIndexedLinear_85779086836295
// MI455X (gfx1250) — compile-verified
//
#include <hip/hip_runtime.h>
#include <hip/hip_bf16.h>

typedef float v2f __attribute__((ext_vector_type(2)));
typedef float v8f __attribute__((ext_vector_type(8)));

namespace {
constexpr int Ud = 512;   // U (K dimension)
constexpr int Vd = 512;   // V (N dimension)
constexpr int BM = 128;   // rows per block
constexpr int BN = 64;    // cols per block
constexpr int BK = 32;    // K chunk staged in LDS
constexpr int XS = 36;    // x LDS row stride (conflict-free, 16B aligned)
constexpr int WS = 72;    // w LDS row stride (half-waves hit disjoint banks)
constexpr int XBUF = BM * XS;   // floats per x buffer
constexpr int WBUF = BK * WS;   // floats per w buffer
}

// LDS aperture keeps the LDS offset in addr[31:0]; truncate generic pointer.
__device__ __forceinline__ uint32_t lds_off(const void* p) {
    return (uint32_t)(uintptr_t)p;
}

// Async global -> LDS copy, 16B per lane, tracked by ASYNCcnt (ISA 08 §4).
__device__ __forceinline__ void async_copy_b128(uint32_t lds_addr, const float* gptr) {
    asm volatile("global_load_async_to_lds_b128 %0, %1, off"
                 :: "v"(lds_addr), "v"((uint64_t)(uintptr_t)gptr)
                 : "memory");
}

__device__ __forceinline__ void wait_asynccnt0() {
    asm volatile("s_wait_asynccnt 0" ::: "memory");
}

__device__ __forceinline__ v8f wmma4(v2f a, v2f b, v8f c) {
    return __builtin_amdgcn_wmma_f32_16x16x4_f32(
        false, a, false, b, (short)0, c, false, false);
}

__global__ __launch_bounds__(256) void sp_indexed_linear_wmma(
    const float* __restrict__ w,      // [C, U*V]
    const float* __restrict__ x,      // [Z, U]
    const int*   __restrict__ idx,    // [Z], sorted
    const float* __restrict__ coeff,  // [1]
    float* __restrict__ y)            // [Z, V]
{
    __shared__ float xs[2 * XBUF];    // double-buffered x tile
    __shared__ float ws[2 * WBUF];    // double-buffered W tile
    __shared__ int   sidx[BM];

    const int tid  = threadIdx.x;
    const int lane = tid & 31;
    const int wave = tid >> 5;
    const int wm   = wave & 3;   // M sub-tile (32 rows per wave)
    const int wn   = wave >> 2;  // N sub-tile (32 cols per wave)

    const int rowBase = blockIdx.y * BM;
    const int colBase = blockIdx.x * BN;

    if (tid < BM) sidx[tid] = idx[rowBase + tid];
    __syncthreads();

    const int gmin = idx[rowBase];            // uniform (idx sorted)
    const int gmax = idx[rowBase + BM - 1];   // uniform

    v8f acc00 = {}, acc01 = {}, acc10 = {}, acc11 = {};

    // staging coordinates: 256 lanes x 16B -> x: 4 rows/lane, w: 2 rows/lane
    const int xr = tid >> 3;          // x rows xr + {0,32,64,96}
    const int xk = (tid & 7) * 4;
    const int wr = tid >> 4;          // w rows wr, wr+16
    const int wc = (tid & 15) * 4;

    uint32_t xl[4];
    const float* xg[4];
    #pragma unroll
    for (int j = 0; j < 4; ++j) {
        const int row = xr + j * 32;
        xl[j] = lds_off(xs) + (uint32_t)((row * XS + xk) * 4);
        xg[j] = x + (size_t)(rowBase + row) * Ud + xk;
    }
    uint32_t wl[2];
    #pragma unroll
    for (int j = 0; j < 2; ++j)
        wl[j] = lds_off(ws) + (uint32_t)(((wr + j * 16) * WS + wc) * 4);

    // WMMA fragment coordinates (V_WMMA_F32_16X16X4_F32 layout)
    const int arow0 = wm * 32 + (lane & 15);
    const int arow1 = arow0 + 16;
    const int kk    = (lane >> 4) * 2;
    const int bcol  = wn * 32 + (lane & 15);

    auto prefetch = [&](uint32_t bufX, uint32_t bufW, int kb,
                        const float* __restrict__ wg) {
        #pragma unroll
        for (int j = 0; j < 4; ++j)
            async_copy_b128(xl[j] + bufX, xg[j] + kb);
        async_copy_b128(wl[0] + bufW, wg + (size_t)(kb + wr)      * Vd);
        async_copy_b128(wl[1] + bufW, wg + (size_t)(kb + wr + 16) * Vd);
    };

    auto compute = [&](const float* __restrict__ xsb,
                       const float* __restrict__ wsb,
                       float m0, float m1, bool masked) {
        #pragma unroll
        for (int k = 0; k < BK; k += 4) {
            v2f a0, a1, b0, b1;
            a0.x = xsb[arow0 * XS + k + kk];
            a0.y = xsb[arow0 * XS + k + kk + 1];
            a1.x = xsb[arow1 * XS + k + kk];
            a1.y = xsb[arow1 * XS + k + kk + 1];
            if (masked) {
                a0.x *= m0; a0.y *= m0;
                a1.x *= m1; a1.y *= m1;
            }
            b0.x = wsb[(k + kk)     * WS + bcol];
            b0.y = wsb[(k + kk + 1) * WS + bcol];
            b1.x = wsb[(k + kk)     * WS + bcol + 16];
            b1.y = wsb[(k + kk + 1) * WS + bcol + 16];
            acc00 = wmma4(a0, b0, acc00);
            acc01 = wmma4(a0, b1, acc01);
            acc10 = wmma4(a1, b0, acc10);
            acc11 = wmma4(a1, b1, acc11);
        }
    };

    auto run_group = [&](int g, bool masked) {
        const float* __restrict__ wg = w + (size_t)g * Ud * Vd + colBase + wc;
        const float m0 = (sidx[arow0] == g) ? 1.f : 0.f;
        const float m1 = (sidx[arow1] == g) ? 1.f : 0.f;

        prefetch(0, 0, 0, wg);
        wait_asynccnt0();
        __syncthreads();

        for (int kb = 0; kb < Ud; kb += 2 * BK) {
            // buf0 holds chunk kb; kb+BK < Ud always (Ud/BK even)
            prefetch(XBUF * 4, WBUF * 4, kb + BK, wg);
            compute(xs, ws, m0, m1, masked);
            wait_asynccnt0();
            __syncthreads();

            const bool more2 = (kb + 2 * BK) < Ud;
            if (more2) prefetch(0, 0, kb + 2 * BK, wg);
            compute(xs + XBUF, ws + WBUF, m0, m1, masked);
            if (more2) wait_asynccnt0();
            __syncthreads();
        }
    };

    if (gmin == gmax) {
        run_group(gmin, false);                       // common case: no masking
    } else {
        for (int g = gmin; g <= gmax; ++g)            // <=7 boundary blocks total
            run_group(g, true);
    }

    // ---- epilogue ----
    const float cf = coeff[0];
    const int rowTop = rowBase + wm * 32 + ((lane >> 4) << 3);  // +8 for upper half-wave
    #pragma unroll
    for (int i = 0; i < 8; ++i) {
        const int r0 = rowTop + i;
        const int r1 = rowTop + 16 + i;
        y[(size_t)r0 * Vd + colBase + bcol]      = acc00[i] * cf;
        y[(size_t)r0 * Vd + colBase + bcol + 16] = acc01[i] * cf;
        y[(size_t)r1 * Vd + colBase + bcol]      = acc10[i] * cf;
        y[(size_t)r1 * Vd + colBase + bcol + 16] = acc11[i] * cf;
    }
}

extern "C" void kernel_launch(void* const* d_in, const int* in_sizes, int n_in,
                              void* d_out, int out_size, void* d_ws, size_t ws_size,
                              hipStream_t stream) {
    const float* w     = (const float*)d_in[0];   // [C, U*V]
    const float* x     = (const float*)d_in[1];   // [Z, U]
    const int*   idx   = (const int*)d_in[2];     // [Z]
    const float* coeff = (const float*)d_in[3];   // [1]
    float* y = (float*)d_out;                     // [Z, V]

    const int Z = in_sizes[1] / Ud;
    dim3 grid(Vd / BN, Z / BM);
    sp_indexed_linear_wmma<<<grid, 256, 0, stream>>>(w, x, idx, coeff, y);
}